// VarianceAdaptor_56375740727384
// MI455X (gfx1250) — compile-verified
//
#include <hip/hip_runtime.h>
#include <hip/hip_bf16.h>
#include <math.h>
#include <stdint.h>

// ---------------- problem geometry (fixed by setup_inputs) ----------------
#define B_   32
#define T_   256
#define D_   256
#define MEL_ 80
#define ML_  2048
#define MROWS (B_ * T_)          // 8192 GEMM rows

typedef float v2f __attribute__((ext_vector_type(2)));
typedef float v8f __attribute__((ext_vector_type(8)));

// D = A(16x4,f32) * B(4x16,f32) + C(16x16,f32)  -> v_wmma_f32_16x16x4_f32
__device__ __forceinline__ v8f wmma4(v2f a, v2f b, v8f c) {
  return __builtin_amdgcn_wmma_f32_16x16x4_f32(
      /*neg_a=*/false, a, /*neg_b=*/false, b,
      /*c_mod=*/(short)0, c, /*reuse_a=*/false, /*reuse_b=*/false);
}

// Async global -> LDS DMA, 16B per lane (tracked by ASYNCcnt).
// lds_addr: segment-relative LDS byte address (low 32 bits of generic ptr).
__device__ __forceinline__ void async_b128(uint32_t lds_addr, const float* gptr) {
  asm volatile("global_load_async_to_lds_b128 %0, %1, off"
               :: "v"(lds_addr), "v"(gptr) : "memory");
}
__device__ __forceinline__ void wait_async0() {
  asm volatile("s_wait_asynccnt 0x0" ::: "memory");
}
__device__ __forceinline__ uint32_t lds_off(const void* p) {
  return (uint32_t)(uintptr_t)p;   // addr[31:0] = LDS segment offset
}

// ---------------- K0: per-batch cumsum of durations + mel_len -------------
__global__ void va_cumsum_kernel(const int* __restrict__ dur,
                                 const int* __restrict__ max_len_p,
                                 int* __restrict__ cum,
                                 int* __restrict__ mellen) {
  int b = blockIdx.x;
  if (threadIdx.x == 0) {
    int ml = max_len_p[0];
    int s = 0;
    for (int t = 0; t < T_; ++t) { s += dur[b * T_ + t]; cum[b * T_ + t] = s; }
    mellen[b] = s < ml ? s : ml;
  }
}

// ---------------- K1/K2: conv1d(k=3) + bias + ReLU + LayerNorm ------------
// One workgroup (16 waves) computes a 16-row x 256-col tile: GEMM M=16,
// N=256, K=768 (im2col over 3 taps), then row-wise LN in LDS.
__global__ __launch_bounds__(512)
void va_conv_ln_kernel(const float* __restrict__ xin,   // [B,T,D]
                       const float* __restrict__ w,     // [3,D,D] row-major (k,cin,cout)
                       const float* __restrict__ bias,  // [D]
                       const float* __restrict__ gamma, // [D]
                       const float* __restrict__ beta,  // [D]
                       float* __restrict__ hout) {      // [B,T,D]
  __shared__ float xs[18 * 256];   // rows t0-1 .. t0+16 (zero padded)
  __shared__ float hs[16 * 256];   // post-ReLU tile for LN

  const int tileM = blockIdx.x;          // 0..511
  const int b     = tileM >> 4;          // 16 tiles per batch
  const int t0    = (tileM & 15) << 4;
  const int tid   = threadIdx.x;

  // halo rows (t0-1, t0+16) -> xs rows 0 and 17, guarded + zero padded
  for (int i = tid; i < 2 * 256; i += 512) {
    int r = (i >> 8) * 17;               // 0 or 17
    int c = i & 255;
    int t = t0 + r - 1;
    xs[r * 256 + c] = (t >= 0 && t < T_) ? xin[((size_t)b * T_ + t) * D_ + c] : 0.f;
  }
  // interior 16 rows (always in-range): async global->LDS DMA, 16KB
  {
    const float* gbase = &xin[((size_t)b * T_ + t0) * D_];
    for (int i = tid; i < 1024; i += 512) {            // 1024 x float4
      async_b128(lds_off(&xs[256 + i * 4]), gbase + i * 4);
    }
    wait_async0();
  }
  __syncthreads();

  const int wave = tid >> 5;             // 16 waves -> 16 N-tiles
  const int lane = tid & 31;
  const int half = lane >> 4;            // K sub-pair select
  const int lid  = lane & 15;            // row (A) / col (B,C)
  const int n0   = wave << 4;

  v8f acc = {};
  for (int kk = 0; kk < 3; ++kk) {
    const float* xrow  = &xs[(lid + kk) * 256];       // A row = lid
    const float* wbase = &w[(size_t)kk * 256 * 256];
    for (int c0 = 0; c0 < 256; c0 += 4) {
      v2f a;
      a.x = xrow[c0 + 2 * half];
      a.y = xrow[c0 + 2 * half + 1];
      v2f bb;
      bb.x = wbase[(size_t)(c0 + 2 * half) * 256 + n0 + lid];
      bb.y = wbase[(size_t)(c0 + 2 * half + 1) * 256 + n0 + lid];
      acc = wmma4(a, bb, acc);
    }
  }

  // epilogue: bias + ReLU into LDS (documented C layout: row = r + 8*half)
  for (int r = 0; r < 8; ++r) {
    int row = r + 8 * half;
    int col = n0 + lid;
    float v = acc[r] + bias[col];
    hs[row * 256 + col] = v > 0.f ? v : 0.f;
  }
  __syncthreads();

  // LayerNorm: wave w normalizes row w (wave32 shuffle reduction)
  {
    int row = wave;
    float s = 0.f, s2 = 0.f;
    for (int c = lane; c < 256; c += 32) {
      float v = hs[row * 256 + c];
      s += v; s2 += v * v;
    }
    for (int off = 16; off > 0; off >>= 1) {
      s  += __shfl_xor(s, off, 32);
      s2 += __shfl_xor(s2, off, 32);
    }
    float mean = s * (1.f / 256.f);
    float var  = s2 * (1.f / 256.f) - mean * mean;
    float inv  = rsqrtf(var + 1e-5f);
    size_t obase = ((size_t)b * T_ + t0 + row) * D_;
    for (int c = lane; c < 256; c += 32) {
      hout[obase + c] = (hs[row * 256 + c] - mean) * inv * gamma[c] + beta[c];
    }
  }
}

// ---------------- K3: linear projection D->1 + mask -----------------------
__global__ __launch_bounds__(256)
void va_proj_kernel(const float* __restrict__ h2, const float* __restrict__ lw,
                    const float* __restrict__ lb_p,
                    const unsigned char* __restrict__ src_mask,
                    float* __restrict__ outp) {
  int row  = blockIdx.x * 8 + (threadIdx.x >> 5);
  int lane = threadIdx.x & 31;
  if (row >= MROWS) return;
  float s = 0.f;
  for (int c = lane; c < 256; c += 32) s += h2[(size_t)row * 256 + c] * lw[c];
  for (int off = 16; off > 0; off >>= 1) s += __shfl_xor(s, off, 32);
  if (lane == 0) {
    float v = s + lb_p[0];
    outp[row] = src_mask[row] ? 0.f : v;
  }
}

// ---------------- K4: fused VAE (5 GEMMs + reparametrization) -------------
// Per 16-row tile: text_mu/text_logvar (K=257 = 256 WMMA + rank-1 duration
// column), mu/logvar (K=80), elementwise z, then vae_emb = z @ up_w (K=256).
__global__ __launch_bounds__(512)
void va_vae_kernel(const float* __restrict__ x,      // [B,T,D]
                   const float* __restrict__ mm,     // [B,T,MEL]
                   const float* __restrict__ eps,    // [B,T,D]
                   const int*   __restrict__ dur,    // [B,T]
                   const float* __restrict__ mu_w,  const float* __restrict__ mu_b,
                   const float* __restrict__ lv_w,  const float* __restrict__ lv_b,
                   const float* __restrict__ up_w,  const float* __restrict__ up_b,
                   const float* __restrict__ tmu_w, const float* __restrict__ tmu_b,
                   const float* __restrict__ tlv_w, const float* __restrict__ tlv_b,
                   float* __restrict__ out_mu,  float* __restrict__ out_lv,
                   float* __restrict__ out_tmu, float* __restrict__ out_tlv,
                   float* __restrict__ vae_emb) {
  __shared__ float xs[16 * 256];
  __shared__ float ms[16 * MEL_];
  __shared__ float ds[16];
  __shared__ float zs[16 * 256];

  const int m0  = blockIdx.x * 16;
  const int tid = threadIdx.x;

  // encoder tile: async global->LDS DMA (contiguous 16KB, no transform)
  for (int i = tid; i < 1024; i += 512) {              // 1024 x float4
    async_b128(lds_off(&xs[i * 4]), &x[(size_t)m0 * 256 + i * 4]);
  }
  // mel tile needs nan_to_num -> through VGPRs
  for (int i = tid; i < 16 * MEL_; i += 512) {
    float v = mm[(size_t)m0 * MEL_ + i];
    ms[i] = isnan(v) ? 0.f : v;
  }
  if (tid < 16) ds[tid] = (float)dur[m0 + tid];
  wait_async0();
  __syncthreads();

  const int wave = tid >> 5, lane = tid & 31;
  const int half = lane >> 4, lid = lane & 15;
  const int n0   = wave << 4;
  const int col  = n0 + lid;

  v8f c_tmu = {}, c_tlv = {}, c_mu = {}, c_lv = {};

  // text posterior over encoder part (weight rows 1..256); A shared per step
  for (int c0 = 0; c0 < 256; c0 += 4) {
    v2f a;
    a.x = xs[lid * 256 + c0 + 2 * half];
    a.y = xs[lid * 256 + c0 + 2 * half + 1];
    v2f bt;
    bt.x = tmu_w[(size_t)(1 + c0 + 2 * half) * 256 + col];
    bt.y = tmu_w[(size_t)(2 + c0 + 2 * half) * 256 + col];
    c_tmu = wmma4(a, bt, c_tmu);
    v2f bl;
    bl.x = tlv_w[(size_t)(1 + c0 + 2 * half) * 256 + col];
    bl.y = tlv_w[(size_t)(2 + c0 + 2 * half) * 256 + col];
    c_tlv = wmma4(a, bl, c_tlv);
  }
  // mel posterior, K=80
  for (int c0 = 0; c0 < MEL_; c0 += 4) {
    v2f a;
    a.x = ms[lid * MEL_ + c0 + 2 * half];
    a.y = ms[lid * MEL_ + c0 + 2 * half + 1];
    v2f bm;
    bm.x = mu_w[(size_t)(c0 + 2 * half) * 256 + col];
    bm.y = mu_w[(size_t)(c0 + 2 * half + 1) * 256 + col];
    c_mu = wmma4(a, bm, c_mu);
    v2f bv;
    bv.x = lv_w[(size_t)(c0 + 2 * half) * 256 + col];
    bv.y = lv_w[(size_t)(c0 + 2 * half + 1) * 256 + col];
    c_lv = wmma4(a, bv, c_lv);
  }

  // epilogue: rank-1 duration column + biases + reparametrize -> z in LDS
  const float tw0 = tmu_w[col];   // weight row 0 (duration feature)
  const float lw0 = tlv_w[col];
  for (int r = 0; r < 8; ++r) {
    int row = r + 8 * half;
    size_t gidx = (size_t)(m0 + row) * 256 + col;
    float d   = ds[row];
    float tmu = c_tmu[r] + d * tw0 + tmu_b[col];
    float tlv = c_tlv[r] + d * lw0 + tlv_b[col];
    float mu  = c_mu[r] + mu_b[col];
    float lv  = c_lv[r] + lv_b[col];
    out_tmu[gidx] = tmu;
    out_tlv[gidx] = tlv;
    out_mu[gidx]  = mu;
    out_lv[gidx]  = lv;
    float tstd = __expf(0.5f * tlv);
    float sstd = __expf(0.5f * lv);
    float z = (eps[gidx] * tstd + tmu) * sstd + mu;
    zs[row * 256 + col] = z;
  }
  __syncthreads();

  // up-projection: vae_emb = z @ up_w + up_b, K=256 from LDS z-tile
  v8f c_up = {};
  for (int c0 = 0; c0 < 256; c0 += 4) {
    v2f a;
    a.x = zs[lid * 256 + c0 + 2 * half];
    a.y = zs[lid * 256 + c0 + 2 * half + 1];
    v2f bu;
    bu.x = up_w[(size_t)(c0 + 2 * half) * 256 + col];
    bu.y = up_w[(size_t)(c0 + 2 * half + 1) * 256 + col];
    c_up = wmma4(a, bu, c_up);
  }
  for (int r = 0; r < 8; ++r) {
    int row = r + 8 * half;
    vae_emb[(size_t)(m0 + row) * 256 + col] = c_up[r] + up_b[col];
  }
}

// ---------------- K5: length regulate (gather) + add, coalesced float4 ----
// Block = 4 frames of one batch; cum[b,:] staged in LDS for the searches.
__global__ __launch_bounds__(256)
void va_regulate_kernel(const float* __restrict__ x, const float* __restrict__ vae,
                        const int* __restrict__ cum, const int* __restrict__ mellen,
                        float* __restrict__ out) {
  __shared__ int cs[T_];
  __shared__ int ml_s;
  int fi = threadIdx.x >> 6;                 // 4 frames per block
  int c4 = threadIdx.x & 63;                 // 64 float4 per 256-ch frame
  int frame_global = blockIdx.x * 4 + fi;    // 0 .. B*ML-1
  int b = frame_global >> 11;
  int f = frame_global & (ML_ - 1);

  if (threadIdx.x < T_) cs[threadIdx.x] = cum[b * T_ + threadIdx.x];
  if (threadIdx.x == 0) ml_s = mellen[b];
  __syncthreads();

  int lo = 0, hi = T_;                       // searchsorted(cum, f, 'right')
  while (lo < hi) { int mid = (lo + hi) >> 1; if (cs[mid] <= f) lo = mid + 1; else hi = mid; }
  int idx = lo < T_ - 1 ? lo : T_ - 1;
  bool valid = f < ml_s;

  const float4* xp = (const float4*)&x[((size_t)b * T_ + idx) * 256];
  const float4* vp = (const float4*)&vae[((size_t)b * T_ + idx) * 256];
  float4 xv = xp[c4], vv = vp[c4];
  float4 o;
  if (valid) { o.x = xv.x + vv.x; o.y = xv.y + vv.y; o.z = xv.z + vv.z; o.w = xv.w + vv.w; }
  else       { o.x = o.y = o.z = o.w = 0.f; }
  ((float4*)&out[((size_t)b * ML_ + f) * 256])[c4] = o;
}

// ---------------- K6: small tail outputs ----------------------------------
__global__ void va_tail_kernel(const int* __restrict__ dur, const int* __restrict__ mellen,
                               const unsigned char* __restrict__ mel_mask,
                               float* __restrict__ out_dur, float* __restrict__ out_mellen,
                               float* __restrict__ out_melmask) {
  int i = blockIdx.x * 256 + threadIdx.x;
  if (i < B_ * T_) out_dur[i] = (float)dur[i];
  if (i < B_)      out_mellen[i] = (float)mellen[i];
  if (i < B_ * ML_) out_melmask[i] = mel_mask[i] ? 1.f : 0.f;
}

// ---------------- launcher ------------------------------------------------
extern "C" void kernel_launch(void* const* d_in, const int* in_sizes, int n_in,
                              void* d_out, int out_size, void* d_ws, size_t ws_size,
                              hipStream_t stream) {
  (void)in_sizes; (void)n_in; (void)out_size; (void)ws_size;
  // inputs (setup_inputs order)
  const float* x          = (const float*)d_in[0];
  const unsigned char* sm = (const unsigned char*)d_in[1];
  const unsigned char* mmk= (const unsigned char*)d_in[2];
  const int*   dur        = (const int*)d_in[3];
  const float* mean_mels  = (const float*)d_in[4];
  const float* text_eps   = (const float*)d_in[5];
  const int*   max_len_p  = (const int*)d_in[6];
  const float* dp_w1 = (const float*)d_in[7];
  const float* dp_b1 = (const float*)d_in[8];
  const float* dp_g1 = (const float*)d_in[9];
  const float* dp_e1 = (const float*)d_in[10];
  const float* dp_w2 = (const float*)d_in[11];
  const float* dp_b2 = (const float*)d_in[12];
  const float* dp_g2 = (const float*)d_in[13];
  const float* dp_e2 = (const float*)d_in[14];
  const float* dp_lw = (const float*)d_in[15];
  const float* dp_lb = (const float*)d_in[16];
  const float* mu_w  = (const float*)d_in[17];
  const float* mu_b  = (const float*)d_in[18];
  const float* lv_w  = (const float*)d_in[19];
  const float* lv_b  = (const float*)d_in[20];
  const float* up_w  = (const float*)d_in[21];
  const float* up_b  = (const float*)d_in[22];
  const float* tmu_w = (const float*)d_in[23];
  const float* tmu_b = (const float*)d_in[24];
  const float* tlv_w = (const float*)d_in[25];
  const float* tlv_b = (const float*)d_in[26];

  // output layout (flat, reference return order, all float32)
  float* o_out    = (float*)d_out;                                   // [B,ML,D]
  float* o_mu     = o_out + (size_t)B_ * ML_ * D_;                   // [B,T,D]
  float* o_lv     = o_mu  + (size_t)MROWS * D_;
  float* o_tmu    = o_lv  + (size_t)MROWS * D_;
  float* o_tlv    = o_tmu + (size_t)MROWS * D_;
  float* o_logdur = o_tlv + (size_t)MROWS * D_;                      // [B,T]
  float* o_dur    = o_logdur + MROWS;                                // [B,T]
  float* o_mellen = o_dur + MROWS;                                   // [B]
  float* o_melmsk = o_mellen + B_;                                   // [B,ML]

  // workspace layout
  char* ws = (char*)d_ws;
  float* h1      = (float*)ws;                                  // 8 MB
  float* h2      = h1 + (size_t)MROWS * D_;                     // 8 MB
  float* vae_emb = h2 + (size_t)MROWS * D_;                     // 8 MB
  int*   cum     = (int*)(vae_emb + (size_t)MROWS * D_);        // 32 KB
  int*   mellen  = cum + MROWS;                                 // 128 B

  // K0: cumsum + mel_len
  va_cumsum_kernel<<<B_, 32, 0, stream>>>(dur, max_len_p, cum, mellen);
  // K1/K2: duration-predictor convs (WMMA f32 GEMM + fused ReLU/LN)
  va_conv_ln_kernel<<<MROWS / 16, 512, 0, stream>>>(x,  dp_w1, dp_b1, dp_g1, dp_e1, h1);
  va_conv_ln_kernel<<<MROWS / 16, 512, 0, stream>>>(h1, dp_w2, dp_b2, dp_g2, dp_e2, h2);
  // K3: projection + mask
  va_proj_kernel<<<MROWS / 8, 256, 0, stream>>>(h2, dp_lw, dp_lb, sm, o_logdur);
  // K4: fused VAE (5 WMMA GEMM chains + reparametrization)
  va_vae_kernel<<<MROWS / 16, 512, 0, stream>>>(x, mean_mels, text_eps, dur,
                                                mu_w, mu_b, lv_w, lv_b, up_w, up_b,
                                                tmu_w, tmu_b, tlv_w, tlv_b,
                                                o_mu, o_lv, o_tmu, o_tlv, vae_emb);
  // K5: length regulate + add (HBM-bound phase, coalesced float4)
  va_regulate_kernel<<<(B_ * ML_) / 4, 256, 0, stream>>>(x, vae_emb, cum, mellen, o_out);
  // K6: tail scalars / masks
  va_tail_kernel<<<(B_ * ML_ + 255) / 256, 256, 0, stream>>>(dur, mellen, mmk,
                                                             o_dur, o_mellen, o_melmsk);
}